// GraphSToV_57062935495220
// MI455X (gfx1250) — compile-verified
//
#include <hip/hip_runtime.h>

// ---------------------------------------------------------------------------
// Problem constants (from reference): B=4, A=200, F=128, OUT=128, C=3
//   h_i = X @ W[:F], h_j = X @ W[F:]          (X flattened to [800,128])
//   out[b,i,j,c,o] = (h_i[b,i,o] + h_j[b,j,o] + bias[o]) * dist[b,i,j,c]
// Output = 245.8 MB f32 -> store-bandwidth bound (~10.5 us @ 23.3 TB/s).
// ---------------------------------------------------------------------------

typedef __attribute__((ext_vector_type(2))) float v2f;
typedef __attribute__((ext_vector_type(4))) float v4f;
typedef __attribute__((ext_vector_type(8))) float v8f;

#define NBATCH 4
#define NATOM  200
#define NFEAT  128
#define NOUT   128
#define NROWS  (NBATCH * NATOM)          // 800
#define NPAIR  (NBATCH * NATOM * NATOM)  // 160000

// ---------------------------------------------------------------------------
// Stage 1: H[half][row][o] = X[row][:] @ W[half*128 + :][o]   (f32 WMMA)
// One wave -> one 16x16 output tile; K=128 swept in 32 steps of 16x16x4.
//
// Fragment layouts per CDNA5 ISA 7.12.2 (wave32):
//   A (16x4 f32, 2 VGPR/lane):  M = lane&15 ; VGPRr holds K = (lane>=16?2:0)+r
//   B (4x16 f32, 2 VGPR/lane):  N = lane&15 ; VGPRr holds K = (lane>=16?2:0)+r
//   C/D (16x16 f32, 8 VGPR):    N = lane&15 ; VGPRr holds M = r + (lane>=16?8:0)
// ---------------------------------------------------------------------------
__global__ __launch_bounds__(128) void hgemm_f32_wmma(
    const float* __restrict__ X,   // [800,128] row-major
    const float* __restrict__ W,   // [256,128] row-major
    float* __restrict__ H)         // ws: [2][800][128]
{
  const int wid  = blockIdx.x * 4 + (threadIdx.x >> 5);  // 0..799
  const int lane = threadIdx.x & 31;

  const int nt   = wid & 7;           // 8 tiles along OUT
  const int mt   = (wid >> 3) % 50;   // 50 tiles along rows
  const int half = wid / 400;         // 0 -> h_i weights, 1 -> h_j weights

  const int mrow = mt * 16 + (lane & 15);
  const int koff = (lane >> 4) * 2;       // 0 for lanes 0-15, 2 for lanes 16-31
  const int ncol = nt * 16 + (lane & 15);

  const float* __restrict__ Xrow = X + mrow * NFEAT + koff;            // contig float2 per step
  const float* __restrict__ Wh   = W + half * NFEAT * NOUT + ncol;     // column ncol of W-half

  v8f acc = {};
#pragma unroll 4
  for (int k0 = 0; k0 < NFEAT; k0 += 4) {
    v2f a, b;
    a.x = Xrow[k0];
    a.y = Xrow[k0 + 1];
    b.x = Wh[(k0 + koff) * NOUT];
    b.y = Wh[(k0 + koff + 1) * NOUT];
    // 8 args: (neg_a, A, neg_b, B, c_mod, C, reuse_a, reuse_b)
    acc = __builtin_amdgcn_wmma_f32_16x16x4_f32(
        false, a, false, b, (short)0, acc, false, false);
  }

  float* __restrict__ Hout = H + half * (NROWS * NOUT);
  const int nbase = nt * 16 + (lane & 15);
  const int mbase = mt * 16 + (lane >> 4) * 8;
#pragma unroll
  for (int r = 0; r < 8; ++r)
    Hout[(mbase + r) * NOUT + nbase] = acc[r];
}

// ---------------------------------------------------------------------------
// Stage 2: bandwidth-bound broadcast. One wave per (b,i,j) pair; each lane
// owns 4 output channels. 3 non-temporal float4 stores per lane (c=0,1,2).
// h rows are hot in L2/WGP$ (0.8 MB total); NT stores keep the 246 MB output
// stream from thrashing the cache hierarchy.
// ---------------------------------------------------------------------------
__global__ __launch_bounds__(256) void broadcast_mul(
    const float* __restrict__ H,      // [2][800][128]
    const float* __restrict__ dist,   // [4,200,200,3]
    const float* __restrict__ bias,   // [128]
    float* __restrict__ out)          // [4,200,200,3,128]
{
  const int p    = blockIdx.x * 8 + (threadIdx.x >> 5);  // pair id, 0..159999
  const int lane = threadIdx.x & 31;
  const int o4   = lane * 4;

  const int j  = p % NATOM;
  const int bi = p / NATOM;           // = b*NATOM + i

  const float* __restrict__ hi = H + (size_t)bi * NOUT;
  const float* __restrict__ hj = H + (size_t)(NROWS + (bi / NATOM) * NATOM + j) * NOUT;

  const v4f pv = *(const v4f*)(hi + o4) + *(const v4f*)(hj + o4)
               + *(const v4f*)(bias + o4);

  const float* __restrict__ d = dist + (size_t)p * 3;
  const float d0 = d[0], d1 = d[1], d2 = d[2];

  float* __restrict__ po = out + (size_t)p * (3 * NOUT) + o4;
  __builtin_nontemporal_store(pv * d0, (v4f*)(po));
  __builtin_nontemporal_store(pv * d1, (v4f*)(po + NOUT));
  __builtin_nontemporal_store(pv * d2, (v4f*)(po + 2 * NOUT));
}

// ---------------------------------------------------------------------------
extern "C" void kernel_launch(void* const* d_in, const int* in_sizes, int n_in,
                              void* d_out, int out_size, void* d_ws, size_t ws_size,
                              hipStream_t stream) {
  const float* X    = (const float*)d_in[0];  // scalar_features [4,200,128]
  const float* dist = (const float*)d_in[1];  // distances       [4,200,200,3]
  const float* W    = (const float*)d_in[2];  // weight          [256,128]
  const float* bias = (const float*)d_in[3];  // bias            [128]
  float* H = (float*)d_ws;                    // needs 2*800*128*4 = 819200 B

  // 800 tiles, 4 waves/block -> 200 blocks
  hgemm_f32_wmma<<<200, 128, 0, stream>>>(X, W, H);
  // 160000 pairs, 8 pairs/block -> 20000 blocks
  broadcast_mul<<<NPAIR / 8, 256, 0, stream>>>(H, dist, bias, (float*)d_out);
}